// KinematicModel_87780541595880
// MI455X (gfx1250) — compile-verified
//
#include <hip/hip_runtime.h>
#include <math.h>

typedef __attribute__((ext_vector_type(2))) float v2f;
typedef __attribute__((ext_vector_type(8))) float v8f;

#define NPTS      50000
#define NCANO     8192
#define NTILES    (NCANO / 16)          // 512 column (cano) tiles
#define NROWTILES ((NPTS + 15) / 16)    // 3125 row (point) tiles, exact: 3125*16 = 50000
#define NJ        15                    // E
#define NPOSE     16                    // E+1 world transforms
#define NTIME     32
#define MROW      4                     // row tiles per wave (B-operand reuse)
#define NGROUPS   ((NROWTILES + MROW - 1) / MROW)  // 782
#define APREP_N   (NCANO + 16)          // one padding tile for the pipelined prefetch

// output layout (floats): pc_trans [32][50000][3] | seg [50000] | trans [32][16][4][4]
#define OUT_PC    0
#define OUT_SEG   (NTIME * NPTS * 3)    // 4,800,000
#define OUT_TRANS (OUT_SEG + NPTS)      // 4,850,000

// --------------------------------------------------------------------------
// Kernel 1: pre-pack cano points for the WMMA A-operand.
// A[m][0..3] = {-2cx, -2cy, -2cz, |c|^2}; padded tail is zero.
// --------------------------------------------------------------------------
__global__ void prep_cano_kernel(const float* __restrict__ cano,
                                 float* __restrict__ aprep) {
  int j = blockIdx.x * blockDim.x + threadIdx.x;
  if (j >= APREP_N) return;
  float x = 0.f, y = 0.f, z = 0.f;
  if (j < NCANO) { x = cano[3 * j]; y = cano[3 * j + 1]; z = cano[3 * j + 2]; }
  float n2 = x * x + y * y + z * z;
  float* o = aprep + 4 * j;
  o[0] = -2.f * x; o[1] = -2.f * y; o[2] = -2.f * z; o[3] = n2;
}

// --------------------------------------------------------------------------
// Kernel 2: forward kinematics. One thread per time step (32 total).
// --------------------------------------------------------------------------
__global__ void fk_kernel(const float* __restrict__ axis,
                          const float* __restrict__ moment,
                          const float* __restrict__ theta,
                          float* __restrict__ trans) {
  int t = threadIdx.x;
  if (t >= NTIME) return;
  float W[16] = {1, 0, 0, 0, 0, 1, 0, 0, 0, 0, 1, 0, 0, 0, 0, 1};
  float* tb = trans + t * NPOSE * 16;
#pragma unroll
  for (int i = 0; i < 16; ++i) tb[i] = W[i];  // world[0] = I

  for (int e = 0; e < NJ; ++e) {
    float ax = axis[3 * e], ay = axis[3 * e + 1], az = axis[3 * e + 2];
    float inv = 1.f / (sqrtf(ax * ax + ay * ay + az * az) + 1e-8f);
    float lx = ax * inv, ly = ay * inv, lz = az * inv;
    float mx = moment[3 * e], my = moment[3 * e + 1], mz = moment[3 * e + 2];
    float qx = ly * mz - lz * my;
    float qy = lz * mx - lx * mz;
    float qz = lx * my - ly * mx;
    float th = theta[t * NJ + e];
    float s = sinf(th), c = cosf(th);
    float K[9] = {0.f, -lz, ly, lz, 0.f, -lx, -ly, lx, 0.f};
    float KK[9];
#pragma unroll
    for (int i = 0; i < 3; ++i)
#pragma unroll
      for (int j = 0; j < 3; ++j)
        KK[i * 3 + j] = K[i * 3 + 0] * K[0 * 3 + j] +
                        K[i * 3 + 1] * K[1 * 3 + j] +
                        K[i * 3 + 2] * K[2 * 3 + j];
    float R[9];
#pragma unroll
    for (int i = 0; i < 9; ++i) R[i] = s * K[i] + (1.f - c) * KK[i];
    R[0] += 1.f; R[4] += 1.f; R[8] += 1.f;
    float q0 = qx, q1 = qy, q2 = qz;
    float tv[3];
#pragma unroll
    for (int i = 0; i < 3; ++i) {
      float d0 = (i == 0 ? 1.f : 0.f) - R[i * 3 + 0];
      float d1 = (i == 1 ? 1.f : 0.f) - R[i * 3 + 1];
      float d2 = (i == 2 ? 1.f : 0.f) - R[i * 3 + 2];
      tv[i] = d0 * q0 + d1 * q1 + d2 * q2;
    }
    float G[16] = {R[0], R[1], R[2], tv[0],
                   R[3], R[4], R[5], tv[1],
                   R[6], R[7], R[8], tv[2],
                   0.f,  0.f,  0.f,  1.f};
    float Wn[16];
#pragma unroll
    for (int i = 0; i < 4; ++i)
#pragma unroll
      for (int j = 0; j < 4; ++j)
        Wn[i * 4 + j] = W[i * 4 + 0] * G[0 * 4 + j] +
                        W[i * 4 + 1] * G[1 * 4 + j] +
                        W[i * 4 + 2] * G[2 * 4 + j] +
                        W[i * 4 + 3] * G[3 * 4 + j];
#pragma unroll
    for (int i = 0; i < 16; ++i) W[i] = Wn[i];
    float* wo = tb + (e + 1) * 16;
#pragma unroll
    for (int i = 0; i < 16; ++i) wo[i] = W[i];
  }
}

// --------------------------------------------------------------------------
// Kernel 3: 1-NN via V_WMMA_F32_16X16X4_F32.
// D[m][n] = |c_m|^2 - 2 x_n . c_m  (A = packed cano rows, B cols = [x,y,z,1]).
// A layout (16x4 f32): lanes 0-15 hold K={0,1}, lanes 16-31 hold K={2,3}.
// B layout (4x16):     lanes 0-15 hold K={0,1}, lanes 16-31 hold K={2,3}.
// C layout (16x16):    VGPR v -> M=v (lanes 0-15) / M=v+8 (lanes 16-31), N=lane%16.
// Each wave: 4 row-tiles of points (B reused), streams 512 cano tiles (A).
// Argmin over cano = within-lane over 8 accumulators + one shfl_xor(16) merge.
// --------------------------------------------------------------------------
__global__ __launch_bounds__(256) void knn_wmma_kernel(
    const float* __restrict__ input_pc, const float* __restrict__ aprep,
    const int* __restrict__ seg_part, float* __restrict__ seg_out) {
  const int lane  = threadIdx.x & 31;
  const int wave  = threadIdx.x >> 5;
  const int group = blockIdx.x * 8 + wave;
  if (group >= NGROUPS) return;          // wave-uniform: EXEC stays all-ones
  const int  lp = lane & 15;
  const bool hi = lane >= 16;

  // B operands: lanes 0-15 -> (x, y); lanes 16-31 -> (z, 1)
  v2f Bv[MROW];
#pragma unroll
  for (int r = 0; r < MROW; ++r) {
    int p = (group * MROW + r) * 16 + lp;
    if (p >= NPTS) p = NPTS - 1;         // clamp loads; writes masked later
    const float* xp = input_pc + 3 * p;
    if (!hi) { Bv[r].x = xp[0]; Bv[r].y = xp[1]; }
    else     { Bv[r].x = xp[2]; Bv[r].y = 1.0f; }
  }

  float best[MROW];
  int   bidx[MROW];
#pragma unroll
  for (int r = 0; r < MROW; ++r) { best[r] = 3.4e38f; bidx[r] = 0; }

  const float* abase = aprep + 4 * lp + (hi ? 2 : 0);
  v2f Acur = *(const v2f*)abase;         // software-pipelined A stream
  const int mb = hi ? 8 : 0;

  for (int t = 0; t < NTILES; ++t) {
    v2f Anext = *(const v2f*)(abase + (t + 1) * 64);  // padded: always valid
#pragma unroll
    for (int r = 0; r < MROW; ++r) {
      v8f acc = {};
      acc = __builtin_amdgcn_wmma_f32_16x16x4_f32(
          /*neg_a=*/false, Acur, /*neg_b=*/false, Bv[r],
          /*c_mod=*/(short)0, acc, /*reuse_a=*/false, /*reuse_b=*/false);
      const int base = t * 16 + mb;
#pragma unroll
      for (int i = 0; i < 8; ++i) {
        float d  = acc[i];
        bool  lt = d < best[r];
        best[r] = lt ? d : best[r];
        bidx[r] = lt ? (base + i) : bidx[r];
      }
    }
    Acur = Anext;
  }

  // merge the two lane halves (M=0..7 vs M=8..15), prefer smaller index on tie
#pragma unroll
  for (int r = 0; r < MROW; ++r) {
    float ob = __shfl_xor(best[r], 16, 32);
    int   oi = __shfl_xor(bidx[r], 16, 32);
    bool take = (ob < best[r]) || (ob == best[r] && oi < bidx[r]);
    int  fi   = take ? oi : bidx[r];
    int  rt   = group * MROW + r;
    if (!hi && rt < NROWTILES) {
      seg_out[rt * 16 + lp] = (float)seg_part[fi];
    }
  }
}

// --------------------------------------------------------------------------
// Kernel 4: apply per-point selected transform. Memory-bound (19.2 MB out).
// --------------------------------------------------------------------------
__global__ __launch_bounds__(256) void apply_kernel(
    const float* __restrict__ input_pc, const float* __restrict__ trans,
    const float* __restrict__ seg_f, float* __restrict__ out_pc) {
  int n = blockIdx.x * blockDim.x + threadIdx.x;
  int t = blockIdx.y;
  if (n >= NPTS) return;
  int seg = (int)seg_f[n];
  const float* M = trans + (t * NPOSE + seg) * 16;
  float x = input_pc[3 * n], y = input_pc[3 * n + 1], z = input_pc[3 * n + 2];
  float* o = out_pc + (size_t)t * (NPTS * 3) + 3 * n;
  o[0] = M[0] * x + M[1] * y + M[2]  * z + M[3];
  o[1] = M[4] * x + M[5] * y + M[6]  * z + M[7];
  o[2] = M[8] * x + M[9] * y + M[10] * z + M[11];
}

// --------------------------------------------------------------------------
extern "C" void kernel_launch(void* const* d_in, const int* in_sizes, int n_in,
                              void* d_out, int out_size, void* d_ws, size_t ws_size,
                              hipStream_t stream) {
  const float* input_pc = (const float*)d_in[0];
  const float* cano_pc  = (const float*)d_in[1];
  const int*   seg_part = (const int*)d_in[2];
  const float* axis     = (const float*)d_in[3];
  const float* moment   = (const float*)d_in[4];
  const float* theta    = (const float*)d_in[5];

  float* out       = (float*)d_out;
  float* out_pc    = out + OUT_PC;
  float* out_seg   = out + OUT_SEG;
  float* out_trans = out + OUT_TRANS;
  float* aprep     = (float*)d_ws;  // APREP_N * 4 floats = ~131 KB

  prep_cano_kernel<<<(APREP_N + 255) / 256, 256, 0, stream>>>(cano_pc, aprep);
  fk_kernel<<<1, 32, 0, stream>>>(axis, moment, theta, out_trans);
  knn_wmma_kernel<<<(NGROUPS + 7) / 8, 256, 0, stream>>>(input_pc, aprep,
                                                         seg_part, out_seg);
  apply_kernel<<<dim3((NPTS + 255) / 256, NTIME), 256, 0, stream>>>(
      input_pc, out_trans, out_seg, out_pc);
}